// TFLSHSelfAttention_46188078301319
// MI455X (gfx1250) — compile-verified
//
#include <hip/hip_runtime.h>
#include <hip/hip_bf16.h>

// ---------------------------------------------------------------------------
// Problem constants (fixed by the reference)
// ---------------------------------------------------------------------------
#define BATCH 2
#define TLEN 4096
#define EMB 1024
#define HEADS 8
#define DH 128
#define BH (BATCH * HEADS)   // 16
#define NHASH 8
#define BUCKET 64
#define NBINS 64             // TLEN / BUCKET
#define NROT 256             // NHASH * (NBINS/2)

typedef float v2f __attribute__((ext_vector_type(2)));
typedef float v8f __attribute__((ext_vector_type(8)));

// fp32 WMMA: D(16x16) = A(16x4) * B(4x16) + C   -> v_wmma_f32_16x16x4_f32
__device__ inline v8f wmma_f32x4(v2f a, v2f b, v8f c) {
  return __builtin_amdgcn_wmma_f32_16x16x4_f32(
      /*neg_a=*/false, a, /*neg_b=*/false, b,
      /*c_mod=*/(short)0, c, /*reuse_a=*/false, /*reuse_b=*/false);
}

// ---------------------------------------------------------------------------
// Generic fp32 WMMA GEMM: C(MxN) = A(MxK) * B(KxN)  (row-major A and B)
// Block tile 128x128, 8 waves (256 thr), each wave 32x64 (2x4 WMMA tiles),
// K staged through LDS in steps of 16.  OUT_MODE 0: plain store (+bias);
// OUT_MODE 1: scatter into merged-head layout (BH, TLEN, DH).
// ---------------------------------------------------------------------------
template <int OUT_MODE>
__global__ __launch_bounds__(256) void gemm_wmma_kernel(
    const float* __restrict__ A, const float* __restrict__ Bm,
    float* __restrict__ C, const float* __restrict__ bias,
    int M, int N, int K, int ldc) {
  __shared__ float As[16][128];  // [k][m] (transposed on load)
  __shared__ float Bs[16][128];  // [k][n]

  const int tid = threadIdx.x;
  const int lane = tid & 31, wid = tid >> 5;
  const int half = lane >> 4, l = lane & 15;
  const int mw = (wid >> 1) * 32;  // wave M offset inside block tile
  const int nw = (wid & 1) * 64;   // wave N offset inside block tile
  const int m0 = blockIdx.y * 128;
  const int n0 = blockIdx.x * 128;

  v8f acc[2][4];
#pragma unroll
  for (int i = 0; i < 2; i++)
#pragma unroll
    for (int j = 0; j < 4; j++)
#pragma unroll
      for (int r = 0; r < 8; r++) acc[i][j][r] = 0.0f;

  for (int kb = 0; kb < K; kb += 16) {
#pragma unroll
    for (int q = 0; q < 2; q++) {
      const int idx = tid * 2 + q;  // 0..511
      // A tile: 128 rows x 16 cols, transposed into As[k][m]
      const int arow = idx >> 2;
      const int akq = (idx & 3) * 4;
      const float4 av =
          *(const float4*)(A + (size_t)(m0 + arow) * K + kb + akq);
      As[akq + 0][arow] = av.x;
      As[akq + 1][arow] = av.y;
      As[akq + 2][arow] = av.z;
      As[akq + 3][arow] = av.w;
      // B tile: 16 rows x 128 cols
      const int brow = idx >> 5;
      const int bc = (idx & 31) * 4;
      *(float4*)&Bs[brow][bc] =
          *(const float4*)(Bm + (size_t)(kb + brow) * N + n0 + bc);
    }
    __syncthreads();

    if (kb + 16 < K) {  // gfx1250 global_prefetch_b8 hints for next tiles
      __builtin_prefetch(A + (size_t)(m0 + (tid >> 1)) * K + kb + 16, 0, 1);
      __builtin_prefetch(Bm + (size_t)(kb + 16 + (tid >> 4)) * N + n0, 0, 1);
    }

#pragma unroll
    for (int k0 = 0; k0 < 16; k0 += 4) {
      v2f a[2], b[4];
#pragma unroll
      for (int mi = 0; mi < 2; mi++) {
        a[mi].x = As[k0 + 2 * half + 0][mw + 16 * mi + l];
        a[mi].y = As[k0 + 2 * half + 1][mw + 16 * mi + l];
      }
#pragma unroll
      for (int ni = 0; ni < 4; ni++) {
        b[ni].x = Bs[k0 + 2 * half + 0][nw + 16 * ni + l];
        b[ni].y = Bs[k0 + 2 * half + 1][nw + 16 * ni + l];
      }
#pragma unroll
      for (int mi = 0; mi < 2; mi++)
#pragma unroll
        for (int ni = 0; ni < 4; ni++)
          acc[mi][ni] = wmma_f32x4(a[mi], b[ni], acc[mi][ni]);
    }
    __syncthreads();
  }

#pragma unroll
  for (int mi = 0; mi < 2; mi++) {
#pragma unroll
    for (int ni = 0; ni < 4; ni++) {
#pragma unroll
      for (int r = 0; r < 8; r++) {
        const int gm = m0 + mw + 16 * mi + r + 8 * half;
        const int gn = n0 + nw + 16 * ni + l;
        float v = acc[mi][ni][r];
        if (OUT_MODE == 0) {
          if (bias) v += bias[gn];
          C[(size_t)gm * ldc + gn] = v;
        } else {
          // merged-head scatter: gm = bidx*TLEN + trow, gn = head*DH + d
          const int bidx = gm >> 12;  // / TLEN
          const int trow = gm & (TLEN - 1);
          const int head = gn >> 7;
          const int d = gn & (DH - 1);
          C[(((size_t)(bidx * HEADS + head)) * TLEN + trow) * DH + d] = v;
        }
      }
    }
  }
}

// ---------------------------------------------------------------------------
// bucket = argmax over [s, -s] of the 32 rotation scores per (B, pos, round)
// ---------------------------------------------------------------------------
__global__ __launch_bounds__(256) void argmax_kernel(
    const float* __restrict__ scores, unsigned short* __restrict__ buckets) {
  const int tid = blockIdx.x * 256 + threadIdx.x;
  if (tid >= BH * TLEN * NHASH) return;
  const int r = tid & (NHASH - 1);
  const int pos = (tid >> 3) & (TLEN - 1);
  const int Bi = tid >> 15;
  const float* s = scores + ((size_t)(Bi * TLEN + pos)) * NROT + r * 32;
  float mx = -3.0e38f, mn = 3.0e38f;
  int ai = 0, mi = 0;
#pragma unroll
  for (int i = 0; i < 32; i++) {
    const float v = s[i];
    if (v > mx) { mx = v; ai = i; }
    if (v < mn) { mn = v; mi = i; }
  }
  const int bkt = (mx >= -mn) ? ai : (32 + mi);
  buckets[(((size_t)(Bi * NHASH + r)) << 12) + pos] = (unsigned short)bkt;
}

// ---------------------------------------------------------------------------
// Stable counting sort per (B, round): 4096 items, 64 buckets.
// Equivalent to stable argsort of t*bucket + pos.  One WG per (B, round).
// ---------------------------------------------------------------------------
__global__ __launch_bounds__(64) void sort_kernel(
    const unsigned short* __restrict__ buckets,
    unsigned short* __restrict__ sticker) {
  __shared__ unsigned short bk[TLEN];
  __shared__ int cnt[NBINS];
  __shared__ int pre[NBINS];
  const int tid = threadIdx.x;
  const size_t base = ((size_t)blockIdx.x) << 12;
  cnt[tid] = 0;
  for (int p = tid; p < TLEN; p += 64) bk[p] = buckets[base + p];
  __syncthreads();
  for (int p = tid; p < TLEN; p += 64) atomicAdd(&cnt[bk[p]], 1);
  __syncthreads();
  if (tid == 0) {
    int run = 0;
    for (int j = 0; j < NBINS; j++) { pre[j] = run; run += cnt[j]; }
  }
  __syncthreads();
  int off = pre[tid];  // thread tid owns bucket tid; scan in pos order (stable)
  for (int p = 0; p < TLEN; p++)
    if (bk[p] == tid) sticker[base + (off++)] = (unsigned short)p;
}

// ---------------------------------------------------------------------------
// Per-bin LSH attention. One WG (128 thr, 4 waves) per (B, round, bin).
// LDS: K rows (normalized, 128x132), V rows (128x132), dots (64x132),
// rowScale/wRow (64 each), positions (128 ints).  Both GEMMs via WMMA fp32.
// Output accumulated unnormalized: num += exp(max)*(expvals@V),
// den += sumexp*exp(max), with hw fp32 global atomics.
// ---------------------------------------------------------------------------
#define SROW 132  // padded LDS row stride (floats)

__global__ __launch_bounds__(128) void lsh_attn_kernel(
    const float* __restrict__ qk, const float* __restrict__ vmat,
    const unsigned short* __restrict__ sticker, float* __restrict__ outAcc,
    float* __restrict__ den) {
  extern __shared__ char smem_raw[];
  float* sK = (float*)smem_raw;            // [128][SROW] normalized keys
  float* sV = sK + 128 * SROW;             // [128][SROW] values
  float* sD = sV + 128 * SROW;             // [64][SROW]  dots / expvals
  float* rowScale = sD + 64 * SROW;        // [64] sqrt(|q|^2)*d^-0.5
  float* wRow = rowScale + 64;             // [64] exp(rowmax)
  int* stk = (int*)(wRow + 64);            // [128] token positions (0..63=own)

  const int tid = threadIdx.x;
  const int bin = blockIdx.x;           // 0..8191
  const int Bi = bin >> 9;              // / (NHASH*NBINS)
  const int rem = bin & 511;
  const int r = rem >> 6;
  const int binIdx = rem & 63;
  const int ebin = (binIdx == 0) ? 1 : binIdx - 1;  // look_one_back partner
  const size_t sbase = ((size_t)(Bi * NHASH + r)) << 12;

  {
    const int slot =
        (tid < 64) ? (binIdx * BUCKET + tid) : (ebin * BUCKET + tid - 64);
    stk[tid] = (int)sticker[sbase + slot];
  }
  __syncthreads();

  // ---- gather + l2-normalize keys, gather values (one row per thread) ----
  {
    const int row = tid;
    const int pos = stk[row];
    const float* src = qk + ((size_t)Bi * TLEN + pos) * DH;
    float sq = 0.0f;
#pragma unroll
    for (int c = 0; c < DH; c += 4) {
      const float4 vv = *(const float4*)(src + c);
      sq += vv.x * vv.x + vv.y * vv.y + vv.z * vv.z + vv.w * vv.w;
      *(float4*)&sK[row * SROW + c] = vv;
    }
    const float sqc = fmaxf(sq, 1e-12f);
    const float rinv = rsqrtf(sqc);
    if (row < 64) rowScale[row] = sqrtf(sqc) * 0.08838834764831845f;  // d^-0.5
#pragma unroll
    for (int c = 0; c < DH; c += 4) {
      float4 vv = *(float4*)&sK[row * SROW + c];
      vv.x *= rinv; vv.y *= rinv; vv.z *= rinv; vv.w *= rinv;
      *(float4*)&sK[row * SROW + c] = vv;
    }
    const float* vsrc = vmat + ((size_t)Bi * TLEN + pos) * DH;
#pragma unroll
    for (int c = 0; c < DH; c += 4)
      *(float4*)&sV[row * SROW + c] = *(const float4*)(vsrc + c);
  }
  __syncthreads();

  const int lane = tid & 31, wid = tid >> 5;
  const int half = lane >> 4, l = lane & 15;
  const int m0w = wid * 16;        // each wave owns 16 query rows
  const int rowA = m0w + l;        // this lane's A row (fixed)
  const float rsA = rowScale[rowA];

  // ---- GEMM1: dots(64x128) = (bq*scale) @ bk^T ----
  v8f acc[8];
#pragma unroll
  for (int ni = 0; ni < 8; ni++)
#pragma unroll
    for (int rr = 0; rr < 8; rr++) acc[ni][rr] = 0.0f;

  for (int k0 = 0; k0 < DH; k0 += 4) {
    v2f a;
    a.x = sK[rowA * SROW + k0 + 2 * half + 0] * rsA;
    a.y = sK[rowA * SROW + k0 + 2 * half + 1] * rsA;
#pragma unroll
    for (int ni = 0; ni < 8; ni++) {
      v2f b;  // B[k][n] = sK[key n][feature k]
      b.x = sK[(16 * ni + l) * SROW + k0 + 2 * half + 0];
      b.y = sK[(16 * ni + l) * SROW + k0 + 2 * half + 1];
      acc[ni] = wmma_f32x4(a, b, acc[ni]);
    }
  }
#pragma unroll
  for (int ni = 0; ni < 8; ni++)
#pragma unroll
    for (int rr = 0; rr < 8; rr++)
      sD[(m0w + rr + 8 * half) * SROW + 16 * ni + l] = acc[ni][rr];
  __syncthreads();

  // ---- masked softmax (unnormalized): sD <- exp(dots - rowmax) ----
  if (tid < 64) {
    const int m = tid;
    const int posq = stk[m];
    float mx = -3.0e38f;
    for (int n = 0; n < 128; n++) {
      float vv = sD[m * SROW + n];
      if (stk[n] == posq) vv = -1e5f;  // self-attention mask
      sD[m * SROW + n] = vv;
      mx = fmaxf(mx, vv);
    }
    float s = 0.0f;
    for (int n = 0; n < 128; n++) {
      const float e = __expf(sD[m * SROW + n] - mx);
      sD[m * SROW + n] = e;
      s += e;
    }
    const float wm = __expf(mx);  // exp(logsumexp) = sum * exp(max)
    wRow[m] = wm;
    unsafeAtomicAdd(den + (size_t)Bi * TLEN + posq, s * wm);
  }
  __syncthreads();

  // ---- GEMM2: bo(64x128) = expvals @ bv ----
#pragma unroll
  for (int ni = 0; ni < 8; ni++)
#pragma unroll
    for (int rr = 0; rr < 8; rr++) acc[ni][rr] = 0.0f;

  for (int k0 = 0; k0 < 128; k0 += 4) {
    v2f a;
    a.x = sD[rowA * SROW + k0 + 2 * half + 0];
    a.y = sD[rowA * SROW + k0 + 2 * half + 1];
#pragma unroll
    for (int ni = 0; ni < 8; ni++) {
      v2f b;  // B[k][n] = sV[key k][feature n]
      b.x = sV[(k0 + 2 * half + 0) * SROW + 16 * ni + l];
      b.y = sV[(k0 + 2 * half + 1) * SROW + 16 * ni + l];
      acc[ni] = wmma_f32x4(a, b, acc[ni]);
    }
  }

  // ---- weighted scatter back to original token order ----
#pragma unroll
  for (int ni = 0; ni < 8; ni++) {
#pragma unroll
    for (int rr = 0; rr < 8; rr++) {
      const int m = m0w + rr + 8 * half;
      const int posq = stk[m];
      const float w = wRow[m];
      const int col = 16 * ni + l;
      unsafeAtomicAdd(outAcc + ((size_t)Bi * TLEN + posq) * DH + col,
                      acc[ni][rr] * w);
    }
  }
}

// ---------------------------------------------------------------------------
// Normalize accumulator and un-merge heads: attn(b, t, e)
// ---------------------------------------------------------------------------
__global__ __launch_bounds__(256) void finalize_kernel(
    const float* __restrict__ outAcc, const float* __restrict__ den,
    float* __restrict__ attn) {
  const int i4 = blockIdx.x * 256 + threadIdx.x;
  if (i4 >= BATCH * TLEN * EMB / 4) return;
  const int elem = i4 * 4;
  const int col = elem & (EMB - 1);
  const int trow = (elem >> 10) & (TLEN - 1);
  const int b_ = elem >> 22;
  const int head = col >> 7;
  const int d = col & (DH - 1);
  const size_t srcRow = (size_t)(b_ * HEADS + head) * TLEN + trow;
  float4 v = *(const float4*)(outAcc + srcRow * DH + d);
  const float inv = 1.0f / den[srcRow];
  v.x *= inv; v.y *= inv; v.z *= inv; v.w *= inv;
  *(float4*)(attn + elem) = v;
}

// ---------------------------------------------------------------------------
// Host launcher
// ---------------------------------------------------------------------------
extern "C" void kernel_launch(void* const* d_in, const int* in_sizes, int n_in,
                              void* d_out, int out_size, void* d_ws,
                              size_t ws_size, hipStream_t stream) {
  const float* x = (const float*)d_in[0];     // (2, 4096, 1024)
  const float* w_qk = (const float*)d_in[1];  // (1024, 1024)
  const float* w_v = (const float*)d_in[2];   // (1024, 1024)
  const float* w_out = (const float*)d_in[3]; // (1024, 1024)
  const float* b_out = (const float*)d_in[4]; // (1024,)
  const float* rot = (const float*)d_in[5];   // (128, 8, 32) -> (128, 256)

  char* ws = (char*)d_ws;
  float* d_qk = (float*)(ws + 0);                            // 32 MB
  float* d_v = (float*)(ws + 33554432ull);                   // 32 MB
  float* d_sc = (float*)(ws + 67108864ull);                  // 64 MB
  unsigned short* d_bkt = (unsigned short*)(ws + 134217728ull);  // 1 MB
  unsigned short* d_stk = (unsigned short*)(ws + 135266304ull);  // 1 MB
  float* d_acc = (float*)(ws + 136314880ull);                // 32 MB
  float* d_den = (float*)(ws + 169869312ull);                // 256 KB
  float* d_attn = (float*)(ws + 170131456ull);               // 32 MB

  const dim3 blk(256);

  // qk / v projections with fused head-merge
  gemm_wmma_kernel<1><<<dim3(8, 64), blk, 0, stream>>>(
      x, w_qk, d_qk, nullptr, BATCH * TLEN, EMB, EMB, 0);
  gemm_wmma_kernel<1><<<dim3(8, 64), blk, 0, stream>>>(
      x, w_v, d_v, nullptr, BATCH * TLEN, EMB, EMB, 0);

  // LSH rotation scores: (BH*TLEN, 128) @ (128, 256)
  gemm_wmma_kernel<0><<<dim3(2, 512), blk, 0, stream>>>(
      d_qk, rot, d_sc, nullptr, BH * TLEN, NROT, DH, NROT);

  // bucket assignment
  argmax_kernel<<<(BH * TLEN * NHASH) / 256, 256, 0, stream>>>(d_sc, d_bkt);

  // stable counting sort per (B, round)
  sort_kernel<<<BH * NHASH, 64, 0, stream>>>(d_bkt, d_stk);

  // zero accumulators (graph-capture-safe)
  hipMemsetAsync(d_acc, 0, 33554432ull, stream);
  hipMemsetAsync(d_den, 0, 262144ull, stream);

  // per-bin attention with dynamic LDS (~166 KB; gfx1250 WGP has 320 KB)
  const size_t attn_lds =
      (size_t)(128 * SROW + 128 * SROW + 64 * SROW + 64 + 64) * 4 + 128 * 4;
  hipFuncSetAttribute((const void*)lsh_attn_kernel,
                      hipFuncAttributeMaxDynamicSharedMemorySize,
                      (int)attn_lds);
  lsh_attn_kernel<<<BH * NHASH * NBINS, 128, attn_lds, stream>>>(
      d_qk, d_v, d_stk, d_acc, d_den);

  // normalize + un-merge heads
  finalize_kernel<<<(BATCH * TLEN * EMB / 4) / 256, 256, 0, stream>>>(
      d_acc, d_den, d_attn);

  // output projection with bias
  gemm_wmma_kernel<0><<<dim3(8, 64), blk, 0, stream>>>(
      d_attn, w_out, (float*)d_out, b_out, BATCH * TLEN, EMB, EMB, EMB);
}